// SpatialGatherModule_1580547972873
// MI455X (gfx1250) — compile-verified
//
#include <hip/hip_runtime.h>
#include <cmath>

typedef __attribute__((ext_vector_type(2))) float v2f;
typedef __attribute__((ext_vector_type(8))) float v8f;

#define B_    16
#define C_    512
#define KR    19        // softmax rows / output K dim
#define HW    16384
#define MPAD  32        // KR padded to 2 WMMA M-tiles

#define NT    128       // N columns per block (8 waves * 16)
#define KC    128       // K chunk staged in LDS per iteration
#define KSPLIT 16       // grid.y split-K factor
#define KITER  8        // chunks per block: KSPLIT*KITER*KC == HW
#define LDA   132       // padded LDS row stride (floats): /4 aligned, bank-safe
#define LDB   132

// ---------------- zero output (atomics accumulate into it) ----------------
__global__ void zero_out_kernel(float* __restrict__ out, int n) {
    int i = blockIdx.x * blockDim.x + threadIdx.x;
    if (i < n) out[i] = 0.0f;
}

// ---------------- per-row softmax stats: max and 1/sum(exp) ---------------
__global__ void row_stats_kernel(const float* __restrict__ probs,
                                 float* __restrict__ ws) {
    const int row = blockIdx.x;              // row = b*KR + k, 304 rows
    const float* p = probs + (size_t)row * HW;
    __shared__ float red[256];
    const int tid = threadIdx.x;

    float m = -3.402823466e38f;
    for (int s = tid; s < HW; s += 256) m = fmaxf(m, p[s]);
    red[tid] = m; __syncthreads();
    for (int w = 128; w > 0; w >>= 1) {
        if (tid < w) red[tid] = fmaxf(red[tid], red[tid + w]);
        __syncthreads();
    }
    const float rmax = red[0]; __syncthreads();

    float s = 0.0f;
    for (int i = tid; i < HW; i += 256) s += __expf(p[i] - rmax);
    red[tid] = s; __syncthreads();
    for (int w = 128; w > 0; w >>= 1) {
        if (tid < w) red[tid] += red[tid + w];
        __syncthreads();
    }
    if (tid == 0) {
        ws[row]            = rmax;
        ws[B_ * KR + row]  = 1.0f / red[0];
    }
}

// ---------------- fused softmax-apply + GEMM via V_WMMA_F32_16X16X4_F32 ---
__global__ void __launch_bounds__(256) gemm_ctx_kernel(
    const float* __restrict__ feat,   // [B, C, HW]
    const float* __restrict__ probs,  // [B, KR, HW]
    const float* __restrict__ ws,     // [304] rowmax, [304] invsum
    float* __restrict__ out)          // [B, C, KR]
{
    extern __shared__ float lds[];
    float* lds_a = lds;                    // [MPAD][LDA] attn tile
    float* lds_b = lds + MPAD * LDA;       // [NT][LDB]   feature tile

    const int b      = blockIdx.z;
    const int n_base = blockIdx.x * NT;
    const int tid    = threadIdx.x;
    const int wave   = tid >> 5;
    const int lane   = tid & 31;
    const int l15    = lane & 15;
    const int hi     = lane >> 4;          // 0 or 1
    const int ko     = hi * 2;             // K-pair select per ISA A layout

    v8f acc0 = {0.f,0.f,0.f,0.f,0.f,0.f,0.f,0.f};   // M rows 0..15
    v8f acc1 = {0.f,0.f,0.f,0.f,0.f,0.f,0.f,0.f};   // M rows 16..31

    const float* pb = probs + (size_t)b * KR * HW;
    const float* fb = feat  + (size_t)b * C_ * HW;

    for (int t = 0; t < KITER; ++t) {
        const int s_base = (blockIdx.y * KITER + t) * KC;

        __syncthreads();  // protect LDS rewrite vs previous iter readers

        // Stage attn tile [MPAD x KC]; exp-normalize once per chunk.
        for (int idx = tid; idx < MPAD * KC; idx += 256) {
            const int s = idx & (KC - 1);
            const int m = idx >> 7;        // idx / KC  (KC==128)
            float v = 0.0f;
            if (m < KR) {
                const float rmax = ws[b * KR + m];
                const float isum = ws[B_ * KR + b * KR + m];
                v = __expf(pb[(size_t)m * HW + s_base + s] - rmax) * isum;
            }
            lds_a[m * LDA + s] = v;
        }
        // Stage feature tile [NT x KC] with coalesced float4 loads.
        const float4* f4 = reinterpret_cast<const float4*>(fb);
        for (int idx = tid; idx < NT * (KC / 4); idx += 256) {
            const int s4 = idx & (KC / 4 - 1);
            const int c  = idx >> 5;       // idx / (KC/4)  (KC/4==32)
            const float4 v =
                f4[((size_t)(n_base + c) * HW + s_base) / 4 + s4];
            *reinterpret_cast<float4*>(&lds_b[c * LDB + s4 * 4]) = v;
        }
        __syncthreads();

        // Prefetch next chunk's tiles into cache while this chunk computes
        // (global_prefetch_b8: fire-and-forget, no counters touched).
        if (t + 1 < KITER) {
            const int s_next = s_base + KC;
            // feature tile: NT rows x 512B -> 4 x 128B segments per row
            for (int idx = tid; idx < NT * 4; idx += 256) {
                const int seg = idx & 3;
                const int c   = idx >> 2;
                __builtin_prefetch(
                    &fb[(size_t)(n_base + c) * HW + s_next + seg * 32], 0, 1);
            }
            // probs tile: KR rows x 512B (only first N-tile block bothers;
            // probs is L2-resident anyway)
            if (blockIdx.x == 0) {
                for (int idx = tid; idx < KR * 4; idx += 256) {
                    const int seg = idx & 3;
                    const int m   = idx >> 2;
                    __builtin_prefetch(
                        &pb[(size_t)m * HW + s_next + seg * 32], 0, 1);
                }
            }
        }

        const int n0 = wave * 16;
        for (int kk = 0; kk < KC; kk += 4) {
            // A (16x4 f32): lane holds M=lane%16; VGPR pair K = ko, ko+1
            v2f a0 = *(const v2f*)&lds_a[l15 * LDA + kk + ko];
            v2f a1 = *(const v2f*)&lds_a[(16 + l15) * LDA + kk + ko];
            // B (4x16 f32): lane holds N=lane%16; VGPR pair K = ko, ko+1
            v2f bf = *(const v2f*)&lds_b[(n0 + l15) * LDB + kk + ko];
            acc0 = __builtin_amdgcn_wmma_f32_16x16x4_f32(
                false, a0, false, bf, (short)0, acc0, false, false);
            acc1 = __builtin_amdgcn_wmma_f32_16x16x4_f32(
                false, a1, false, bf, (short)0, acc1, false, false);
        }
    }

    // D layout (f32 16x16): VGPR r -> lanes 0-15: M=r, N=lane;
    //                                 lanes 16-31: M=r+8, N=lane-16.
    const int n = n_base + wave * 16 + l15;
    float* outn = out + (size_t)(b * C_ + n) * KR;
    #pragma unroll
    for (int r = 0; r < 8; ++r) {
        const int m0 = r + hi * 8;
        if (m0 < KR) atomicAdd(&outn[m0], acc0[r]);
        const int m1 = m0 + 16;
        if (m1 < KR) atomicAdd(&outn[m1], acc1[r]);
    }
}

extern "C" void kernel_launch(void* const* d_in, const int* in_sizes, int n_in,
                              void* d_out, int out_size, void* d_ws, size_t ws_size,
                              hipStream_t stream) {
    (void)in_sizes; (void)n_in; (void)ws_size;
    const float* features = (const float*)d_in[0];  // [16,512,128,128]
    const float* probs    = (const float*)d_in[1];  // [16,19,128,128]
    float* out = (float*)d_out;                     // [16,512,19,1]
    float* ws  = (float*)d_ws;                      // 608 floats used

    zero_out_kernel<<<(out_size + 255) / 256, 256, 0, stream>>>(out, out_size);
    row_stats_kernel<<<B_ * KR, 256, 0, stream>>>(probs, ws);

    dim3 grid(C_ / NT, KSPLIT, B_);
    const size_t ldsBytes = (size_t)(MPAD * LDA + NT * LDB) * sizeof(float);
    gemm_ctx_kernel<<<grid, 256, ldsBytes, stream>>>(features, probs, ws, out);
}